// TOI_Pooling_76510547411005
// MI455X (gfx1250) — compile-verified
//
#include <hip/hip_runtime.h>
#include <cstdint>

// Problem constants (match reference)
#define Bb 8
#define Dd 512
#define Tt 1024
#define Ss 4096
#define Pp 8
#define NROW 10              // P+2 output rows per segment
#define CH_PER_BLK 256       // channels staged per block (blockIdx.y selects half)
#define LDS_STRIDE 68        // 64 data dwords + 4 pad dwords -> conflict-free ds_load_2addr
#define LDS_BYTES (CH_PER_BLK * LDS_STRIDE * 4)

typedef float v2f __attribute__((ext_vector_type(2)));
typedef float v8f __attribute__((ext_vector_type(8)));
typedef unsigned int v4u __attribute__((ext_vector_type(4)));
typedef int v8i __attribute__((ext_vector_type(8)));
typedef int v4i __attribute__((ext_vector_type(4)));

__global__ __launch_bounds__(128) void toi_pool_wmma(const float* __restrict__ features,
                                                     const int* __restrict__ tois,
                                                     float* __restrict__ out) {
    extern __shared__ float lds[];

    const int seg  = blockIdx.x;           // b*S + s
    const int half = blockIdx.y;           // which 256-channel half
    const int b    = seg >> 12;            // S = 4096 = 2^12
    const int start = tois[2 * seg + 0];
    const int L     = tois[2 * seg + 1] - start;

    const float* gsrc = features + ((size_t)b * Dd + (size_t)half * CH_PER_BLK) * Tt + start;

    // ---- Stage the 256(ch) x 64(time) f32 window into LDS via TDM, row stride 68 dwords ----
#if __has_builtin(__builtin_amdgcn_tensor_load_to_lds) && __has_builtin(__builtin_amdgcn_s_wait_tensorcnt)
    if ((threadIdx.x >> 5) == 0) {         // one wave issues the TDM op
        const unsigned lds_base = (unsigned)(uintptr_t)&lds[0];
        const unsigned long long ga = (unsigned long long)(uintptr_t)gsrc;
        v4u g0;
        g0.x = 1u;                                                // count=1 (valid), user mode
        g0.y = lds_base;                                          // lds_addr
        g0.z = (unsigned)(ga & 0xFFFFFFFFu);                      // global_addr[31:0]
        g0.w = (unsigned)((ga >> 32) & 0x01FFFFFFu) | (2u << 30); // global_addr[56:32] | type=2
        v8i g1;
        g1[0] = (int)((2u << 16)      // data_size = 4B
                    | (1u << 20)      // pad_enable
                    | (5u << 22)      // pad_interval: 64 dwords
                    | (3u << 25));    // pad_amount: 4 dwords  -> LDS row stride 68
        g1[1] = (int)((unsigned)(Tt & 0xFFFF) << 16);                          // tensor_dim0 lo
        g1[2] = (int)(((unsigned)Tt >> 16) | ((unsigned)(Dd & 0xFFFF) << 16)); // dim0 hi | dim1 lo
        g1[3] = (int)(((unsigned)Dd >> 16) | (64u << 16));                     // dim1 hi | tile_dim0=64
        g1[4] = (int)CH_PER_BLK;                                               // tile_dim1=256, tile_dim2=0
        g1[5] = (int)Tt;                                                       // tensor_dim0_stride = 1024
        g1[6] = 0;
        g1[7] = 0;
        const v4i gz4 = {0, 0, 0, 0};                                          // 2D tensor: groups 2/3 unused
        const v8i gz8 = {0, 0, 0, 0, 0, 0, 0, 0};
        __builtin_amdgcn_tensor_load_to_lds(g0, g1, gz4, gz4, gz8, 0);
        __builtin_amdgcn_s_wait_tensorcnt(0);
    }
    __syncthreads();
#else
    for (int linear = threadIdx.x; linear < CH_PER_BLK * 64; linear += 128) {
        const int d = linear >> 6, j = linear & 63;                            // coalesced over j
        lds[d * LDS_STRIDE + j] = gsrc[(size_t)d * Tt + j];
    }
    __syncthreads();
#endif

    // ---- Per-lane A-operand (weights), documented 16x4 f32 layout ----
    const int lane = threadIdx.x & 31;
    const int wave = threadIdx.x >> 5;
    const int m = lane & 15;               // A row / B,C column within tile
    const int h = lane >> 4;               // K half select (0: K0,K1 ; 1: K2,K3)

    const int Lm2 = L - 2;
    const int mm  = (Lm2 > Pp) ? (Lm2 >> 3) : 1;
    const int rr  = (Lm2 > Pp) ? (Lm2 & 7) : 0;
    const float inv = 1.0f / (float)(mm + rr);

    // Branchless weight construction (pure expression tree -> v_cmp/v_cndmask, no EXEC ladders).
    // Rows 0 / 9 / "exact" mids all pick exactly one sample; "large" mids average [lo, hi).
    const bool isMid = (m >= 1) && (m <= Pp);
    const bool large = isMid && (Lm2 > Pp);
    const bool exact = isMid && (Lm2 >= m) && (Lm2 <= Pp);
    int pick = -1;                                       // -1: matches no k
    pick = (m == 0) ? 0 : pick;                          // first sample
    pick = exact ? m : pick;                             // exact interior pick
    pick = ((m == 9) && (L > 1)) ? (L - 1) : pick;       // last sample
    const int lo = 1 + (m - 1) * mm;                     // averaged window [lo, hi)
    const int hi = 1 + m * mm + rr;

    v2f a[16];
#pragma unroll
    for (int kk = 0; kk < 16; ++kk) {
        const int k0 = kk * 4 + 2 * h;
        const int k1 = k0 + 1;
        a[kk].x = (k0 == pick) ? 1.0f : ((large && (k0 >= lo) && (k0 < hi)) ? inv : 0.0f);
        a[kk].y = (k1 == pick) ? 1.0f : ((large && (k1 >= lo) && (k1 < hi)) ? inv : 0.0f);
    }

    // ---- 4 N-tiles of 16 channels per wave: C(16x16) = W(16x64) x F(64x16) ----
    const size_t outSegBase = (size_t)seg * (NROW * Dd) + (size_t)half * CH_PER_BLK;

#pragma unroll
    for (int t = 0; t < 4; ++t) {
        const int cbase = (wave * 4 + t) * 16;
        v8f c = {0.f, 0.f, 0.f, 0.f, 0.f, 0.f, 0.f, 0.f};
#pragma unroll
        for (int kk = 0; kk < 16; ++kk) {
            const int col = kk * 4 + 2 * h;  // B layout mirrors A: lanes 0-15 K0/K1, 16-31 K2/K3
            const v2f bv = *(const v2f*)&lds[(size_t)(cbase + m) * LDS_STRIDE + col];
            c = __builtin_amdgcn_wmma_f32_16x16x4_f32(false, a[kk], false, bv,
                                                      (short)0, c, false, false);
        }
        // C/D layout: vgpr r -> row r (lanes 0-15), row r+8 (lanes 16-31); col = lane&15.
        const size_t colAddr = outSegBase + (size_t)(cbase + m);
#pragma unroll
        for (int r = 0; r < 8; ++r) {
            const int p = r + h * 8;
            if (p < NROW) out[colAddr + (size_t)p * Dd] = c[r];  // coalesced 64B runs
        }
    }

    // ---- counts tail: cumsum of S per batch, stored as f32 after pooled ----
    if (blockIdx.x == 0 && blockIdx.y == 0 && threadIdx.x < Bb) {
        out[(size_t)Bb * Ss * NROW * Dd + threadIdx.x] = (float)((threadIdx.x + 1) * Ss);
    }
}

extern "C" void kernel_launch(void* const* d_in, const int* in_sizes, int n_in,
                              void* d_out, int out_size, void* d_ws, size_t ws_size,
                              hipStream_t stream) {
    (void)in_sizes; (void)n_in; (void)d_ws; (void)ws_size; (void)out_size;
    const float* features = (const float*)d_in[0];
    const int*   tois     = (const int*)d_in[1];
    float*       out      = (float*)d_out;

    // 69,632 B dynamic LDS (> default 64KB cap on some stacks; gfx1250 WGP has 320KB)
    static_assert(LDS_BYTES == 69632, "lds size");
    (void)hipFuncSetAttribute((const void*)toi_pool_wmma,
                              hipFuncAttributeMaxDynamicSharedMemorySize, LDS_BYTES);

    dim3 grid(Bb * Ss, 2);
    dim3 block(128);
    hipLaunchKernelGGL(toi_pool_wmma, grid, block, LDS_BYTES, stream, features, tois, out);
}